// VectorQuantizer_41248865910805
// MI455X (gfx1250) — compile-verified
//
#include <hip/hip_runtime.h>
#include <hip/hip_bf16.h>

// VQ nearest-codebook: z_e [32,64,64,64] f32, codebook [512,64] f32.
// N = 131072 rows, K = 512 codes, D = 64.
// ~8.6 GFLOP fp32 via V_WMMA_F32_16X16X4_F32; ~67MB moved -> AI ~128 FLOP/B,
// matrix-op bound. fp32 WMMA keeps the argmin numerically faithful (codebook
// entries are +-1/512; bf16/f16 inputs could flip nearest-code decisions).

typedef __attribute__((ext_vector_type(2))) float v2f;
typedef __attribute__((ext_vector_type(8))) float v8f;

#define KCODES 512
#define DDIM   64
#define HWSZ   4096                 // H*W
#define NROWS  (32 * HWSZ)          // 131072
// B-tile LDS layout: bt[c2][code] holds float2 {e[code][2*c2], e[code][2*c2+1]}
// row stride 528 float2 = 1056 dwords; 1056 % 64 == 32 -> the two half-waves
// (c2 = 2kk vs 2kk+1) hit disjoint 32-bank halves: conflict-free ds_load_b64.
#define BTROW  (528 * 2)            // dwords per c2 row

#if defined(__has_builtin)
#if __has_builtin(__builtin_amdgcn_global_load_async_to_lds_b32)
#define HAVE_ASYNC_LDS 1
#endif
#endif

typedef __attribute__((address_space(1))) int gint_t;   // global int
typedef __attribute__((address_space(3))) int lint_t;   // LDS int

__global__ __launch_bounds__(256) void
vq_wmma_kernel(const float* __restrict__ z,      // [B,C,H,W]
               const float* __restrict__ cb,     // [K,D]
               float* __restrict__ zq,           // [N,D] flat (raw reshape)
               int* __restrict__ idx_out)        // [N]
{
    __shared__ __align__(16) float bt[32 * BTROW];   // paired transposed codebook
    __shared__ float cbn[KCODES];                    // ||e_k||^2

    const int tid = threadIdx.x;

    // ---- stage codebook into LDS (transposed scatter into float2 pairs) ----
#if HAVE_ASYNC_LDS
    {
        // Async DMA path: per-lane LDS destination VGPR does the transpose;
        // tracked on ASYNCcnt, bypasses the VMEM->VGPR->DS round trip.
        for (int i = tid; i < KCODES * DDIM; i += 256) {
            int code = i >> 6;          // coalesced: lanes walk c (fast dim of cb)
            int c    = i & 63;
            gint_t* src = (gint_t*)(cb + i);
            lint_t* dst = (lint_t*)&bt[(c >> 1) * BTROW + code * 2 + (c & 1)];
            __builtin_amdgcn_global_load_async_to_lds_b32(src, dst,
                                                          /*offset=*/0, /*cpol=*/0);
        }
        asm volatile("s_wait_asynccnt 0x0" ::: "memory");
    }
#else
    for (int i = tid; i < KCODES * (DDIM / 4); i += 256) {
        int code = i >> 4;
        int c4   = (i & 15) << 2;
        const float4 v = *(const float4*)(cb + code * DDIM + c4);
        bt[((c4 + 0) >> 1) * BTROW + code * 2 + 0] = v.x;
        bt[((c4 + 0) >> 1) * BTROW + code * 2 + 1] = v.y;
        bt[((c4 + 2) >> 1) * BTROW + code * 2 + 0] = v.z;
        bt[((c4 + 2) >> 1) * BTROW + code * 2 + 1] = v.w;
    }
#endif
    __syncthreads();

    // ---- codebook norms from LDS ----
    for (int code = tid; code < KCODES; code += 256) {
        float s = 0.f;
        #pragma unroll
        for (int c2 = 0; c2 < 32; ++c2) {
            v2f e = *(const v2f*)&bt[c2 * BTROW + code * 2];
            s += e.x * e.x + e.y * e.y;
        }
        cbn[code] = s;
    }
    __syncthreads();

    const int wave  = tid >> 5;
    const int lane  = tid & 31;
    const int nlane = lane & 15;      // row/column-within-tile lane
    const int hv    = lane >> 4;      // half-wave (K-pair select in WMMA layout)

    const int n0  = blockIdx.x * 128 + wave * 16;   // this wave's 16 rows
    const int b   = n0 >> 12;                       // 4096 rows per batch image
    const int hw0 = n0 & (HWSZ - 1);
    // z_flat[n][c] = z_e[b][c][hw]; rows are the fast hw axis (coalesced 64B/half-wave)
    const float* zb = z + (size_t)b * DDIM * HWSZ + hw0 + nlane;

    // ---- A tile: 16 rows x 64 dims as 16 K-steps of 16x4 fp32, kept resident ----
    // layout: VGPR0 lanes0-15 K=c+0 / lanes16-31 K=c+2 ; VGPR1 K=c+1 / c+3
    v2f a[16];
    #pragma unroll
    for (int kk = 0; kk < 16; ++kk) {
        int c = 4 * kk + 2 * hv;
        a[kk].x = zb[(size_t)(c + 0) * HWSZ];
        a[kk].y = zb[(size_t)(c + 1) * HWSZ];
    }

    float best[8];
    int   bidx[8];
    #pragma unroll
    for (int v = 0; v < 8; ++v) { best[v] = 3.4e38f; bidx[v] = 0; }

    // ---- 32 column tiles of 16 codes; 16 chained WMMAs per tile (K=64) ----
    #pragma unroll 2
    for (int t = 0; t < 32; ++t) {
        v8f acc = {};
        const int codecol = t * 16 + nlane;
        #pragma unroll
        for (int kk = 0; kk < 16; ++kk) {
            // one conflict-free ds_load_b64 supplies both B VGPRs for this step
            v2f bvec = *(const v2f*)&bt[(2 * kk + hv) * BTROW + codecol * 2];
            acc = __builtin_amdgcn_wmma_f32_16x16x4_f32(
                      /*neg_a=*/false, a[kk], /*neg_b=*/false, bvec,
                      /*c_mod=*/(short)0, acc, /*reuse_a=*/false, /*reuse_b=*/false);
        }
        // distance minus row-constant ||x||^2:  ||e||^2 - 2 x.e
        const float cn = cbn[codecol];
        #pragma unroll
        for (int v = 0; v < 8; ++v) {
            float d = cn - 2.0f * acc[v];
            if (d < best[v]) { best[v] = d; bidx[v] = codecol; }  // ascending -> first-min kept
        }
    }

    // ---- argmin across the 16 lanes of each half-wave (each holds one row set) ----
    #pragma unroll
    for (int v = 0; v < 8; ++v) {
        #pragma unroll
        for (int off = 8; off >= 1; off >>= 1) {
            float ov = __shfl_xor(best[v], off, 16);
            int   oi = __shfl_xor(bidx[v], off, 16);
            if (ov < best[v] || (ov == best[v] && oi < bidx[v])) { best[v] = ov; bidx[v] = oi; }
        }
    }

    // acc VGPR v <-> row (v + 8*hv): lanes 0 and 16 publish rows 0-7 / 8-15
    if (nlane == 0) {
        #pragma unroll
        for (int v = 0; v < 8; ++v) idx_out[n0 + hv * 8 + v] = bidx[v];
    }

    // ---- gather selected codebook rows to output (coalesced 128B, cb is L2-hot) ----
    #pragma unroll
    for (int r = 0; r < 16; ++r) {
        int src = (r < 8) ? 0 : 16;                 // group holding row r after reduction
        int ir  = __shfl(bidx[r & 7], src, 32);
        zq[(size_t)(n0 + r) * DDIM + lane]      = cb[(size_t)ir * DDIM + lane];
        zq[(size_t)(n0 + r) * DDIM + lane + 32] = cb[(size_t)ir * DDIM + lane + 32];
    }
}

extern "C" void kernel_launch(void* const* d_in, const int* in_sizes, int n_in,
                              void* d_out, int out_size, void* d_ws, size_t ws_size,
                              hipStream_t stream) {
    const float* z_e  = (const float*)d_in[0];   // [32,64,64,64]
    const float* cb   = (const float*)d_in[1];   // [512,64]
    float* zq   = (float*)d_out;                                // 8388608 f32
    int*   idxo = (int*)((float*)d_out + (size_t)NROWS * DDIM); // 131072 int32

    dim3 grid(NROWS / 128);   // 1024 blocks, 8 waves * 16 rows each
    dim3 block(256);
    vq_wmma_kernel<<<grid, block, 0, stream>>>(z_e, cb, zq, idxo);
}